// BMERtl_22814866276454
// MI455X (gfx1250) — compile-verified
//
#include <hip/hip_runtime.h>

typedef __attribute__((ext_vector_type(2))) float v2f;
typedef __attribute__((ext_vector_type(4))) float v4f;
typedef __attribute__((ext_vector_type(8))) float v8f;

#define Bsz    8
#define Natoms 100000
#define Ccon   64
#define Kdim   (Natoms * 3)        // 300000 reduction length per batch
#define CHUNKS 125                 // K-chunks per batch
#define KCHUNK (Kdim / CHUNKS)     // 2400, divisible by 8

// workspace layout (float offsets)
#define WS_G    0                               // [B][64][64] Gram accumulators
#define WS_BP   (WS_G   + Bsz * Ccon * Ccon)    // [B][64]  b-vector accumulators
#define WS_LMD  (WS_BP  + Bsz * Ccon)           // [B][64]  solved lambda
#define WS_HDR  (WS_LMD + Bsz * Ccon)           // 33792 floats header
#define WS_WGT  WS_HDR                          // weights: [B][K] (mode0) or [B][N] (mode1)

// --- init, mode 0: zero header + expand 1/mas to per-K array ---------------
__global__ void __launch_bounds__(256)
bme_init_expand(const float* __restrict__ mas, float* __restrict__ ws) {
    int idx = blockIdx.x * blockDim.x + threadIdx.x;    // B*N threads
    if (idx < WS_HDR) ws[idx] = 0.0f;
    if (idx < Bsz * Natoms) {
        int b = idx / Natoms;
        int n = idx - b * Natoms;
        float w = 1.0f / mas[idx];
        float* dst = ws + WS_WGT + (size_t)b * Kdim + 3 * n;
        dst[0] = w; dst[1] = w; dst[2] = w;
    }
}

// --- init, mode 1: zero header + per-atom 1/mas ----------------------------
__global__ void __launch_bounds__(256)
bme_init_compact(const float* __restrict__ mas, float* __restrict__ ws) {
    int idx = blockIdx.x * blockDim.x + threadIdx.x;    // B*N threads
    if (idx < WS_HDR) ws[idx] = 0.0f;
    if (idx < Bsz * Natoms) ws[WS_WGT + idx] = 1.0f / mas[idx];
}

// --- Gram inner loop: 8 K-columns/iter, 4 WMMAs, all-b128 feeds ------------
// K-slot remap: lane-half h, wmma-step t, vgpr j  <->  memory column q+4h+2t+j
// (valid because the Gram reduction is K-order invariant; A and B agree).
template <bool DOB, int WMODE>
__device__ __forceinline__ void
gram_loop(const float* __restrict__ rowA,   // + 4*half already applied
          const float* __restrict__ rowB0,
          const float* __restrict__ rowB1,
          const float* __restrict__ wgt,    // mode0: +4*half applied; mode1: per-atom
          const float* __restrict__ mrow,   // + 4*half applied
          int k0, int kend,
          v8f& acc0, v8f& acc1, float& bacc) {
#pragma unroll 2
    for (int q = k0; q < kend; q += 8) {
        v4f fa = *(const v4f*)(rowA + q);
        v4f wv;
        if (WMODE == 0) {
            wv = *(const v4f*)(wgt + q);
        } else {
            // wgt indexed per atom: column (rowA+q)'s K index is q + 4*half,
            // but rowA already carries +4*half, so recover via pointer math is
            // messy -- instead mode 1 passes wgt pre-offset per-K? No: do /3 here.
            // kq = q + (implicit 4*half folded into separate base) -> caller passes
            // kofs so that true K = q + kofs.
            wv.x = 0.f; wv.y = 0.f; wv.z = 0.f; wv.w = 0.f; // overwritten below
        }
        if (WMODE == 1) {
            // caller guarantees wgt = invm base, and rowA offset == 4*half,
            // true K column for fa.x is q + kofs where kofs encoded in mrow-rowA? 
            // -> handled by caller passing kofs via template-free trick below.
        }
        v4f faw = fa * wv;
        v4f fb0 = *(const v4f*)(rowB0 + q);
        v4f fb1 = *(const v4f*)(rowB1 + q);
        v2f a0;  a0.x  = faw.x; a0.y  = faw.y;
        v2f a1;  a1.x  = faw.z; a1.y  = faw.w;
        v2f b00; b00.x = fb0.x; b00.y = fb0.y;
        v2f b01; b01.x = fb0.z; b01.y = fb0.w;
        v2f b10; b10.x = fb1.x; b10.y = fb1.y;
        v2f b11; b11.x = fb1.z; b11.y = fb1.w;
        acc0 = __builtin_amdgcn_wmma_f32_16x16x4_f32(false, a0, false, b00,
                                                     (short)0, acc0, false, false);
        acc1 = __builtin_amdgcn_wmma_f32_16x16x4_f32(false, a0, false, b10,
                                                     (short)0, acc1, false, false);
        acc0 = __builtin_amdgcn_wmma_f32_16x16x4_f32(false, a1, false, b01,
                                                     (short)0, acc0, false, false);
        acc1 = __builtin_amdgcn_wmma_f32_16x16x4_f32(false, a1, false, b11,
                                                     (short)0, acc1, false, false);
        if (DOB) {
            v4f m = *(const v4f*)(mrow + q);
            bacc += faw.x * m.x + faw.y * m.y + faw.z * m.z + faw.w * m.w;
        }
    }
}

// mode-1 variant with explicit per-column /3 weight lookup
template <bool DOB>
__device__ __forceinline__ void
gram_loop_compact(const float* __restrict__ rowA, const float* __restrict__ rowB0,
                  const float* __restrict__ rowB1, const float* __restrict__ invm,
                  const float* __restrict__ mrow, int kofs, int k0, int kend,
                  v8f& acc0, v8f& acc1, float& bacc) {
#pragma unroll 2
    for (int q = k0; q < kend; q += 8) {
        v4f fa = *(const v4f*)(rowA + q);
        unsigned kq = (unsigned)(q + kofs);
        v4f wv;
        wv.x = invm[kq / 3u];
        wv.y = invm[(kq + 1) / 3u];
        wv.z = invm[(kq + 2) / 3u];
        wv.w = invm[(kq + 3) / 3u];
        v4f faw = fa * wv;
        v4f fb0 = *(const v4f*)(rowB0 + q);
        v4f fb1 = *(const v4f*)(rowB1 + q);
        v2f a0;  a0.x  = faw.x; a0.y  = faw.y;
        v2f a1;  a1.x  = faw.z; a1.y  = faw.w;
        v2f b00; b00.x = fb0.x; b00.y = fb0.y;
        v2f b01; b01.x = fb0.z; b01.y = fb0.w;
        v2f b10; b10.x = fb1.x; b10.y = fb1.y;
        v2f b11; b11.x = fb1.z; b11.y = fb1.w;
        acc0 = __builtin_amdgcn_wmma_f32_16x16x4_f32(false, a0, false, b00,
                                                     (short)0, acc0, false, false);
        acc1 = __builtin_amdgcn_wmma_f32_16x16x4_f32(false, a0, false, b10,
                                                     (short)0, acc1, false, false);
        acc0 = __builtin_amdgcn_wmma_f32_16x16x4_f32(false, a1, false, b01,
                                                     (short)0, acc0, false, false);
        acc1 = __builtin_amdgcn_wmma_f32_16x16x4_f32(false, a1, false, b11,
                                                     (short)0, acc1, false, false);
        if (DOB) {
            v4f m = *(const v4f*)(mrow + q);
            bacc += faw.x * m.x + faw.y * m.y + faw.z * m.z + faw.w * m.w;
        }
    }
}

// --- kernel 2: G = (J M^-1) J^T and b = (J M^-1) p via f32 WMMA ------------
template <int WMODE>
__global__ void __launch_bounds__(256)
bme_gram_kernel(const float* __restrict__ jac, const float* __restrict__ mom,
                float* __restrict__ ws) {
    const int b     = blockIdx.x / CHUNKS;
    const int chunk = blockIdx.x % CHUNKS;
    const int k0    = chunk * KCHUNK;
    const int kend  = k0 + KCHUNK;

    const int lane = threadIdx.x & 31;
    const int wave = threadIdx.x >> 5;
    const int half = lane >> 4;
    const int l15  = lane & 15;
    const int ci   = wave >> 1;        // output row tile 0..3
    const int cj0  = (wave & 1) * 2;   // first of two output col tiles
    const int hofs = 4 * half;

    const float* __restrict__ Jb   = jac + (size_t)b * Ccon * Kdim;
    const float* __restrict__ momb = mom + (size_t)b * Kdim + hofs;

    const float* rowA  = Jb + (size_t)(ci * 16 + l15) * Kdim + hofs;
    const float* rowB0 = Jb + (size_t)(cj0 * 16 + l15) * Kdim + hofs;
    const float* rowB1 = Jb + (size_t)((cj0 + 1) * 16 + l15) * Kdim + hofs;

    v8f acc0 = {};
    v8f acc1 = {};
    float bacc = 0.0f;

    if (WMODE == 0) {
        const float* wgt = ws + WS_WGT + (size_t)b * Kdim + hofs;
        if ((wave & 1) == 0)
            gram_loop<true, 0>(rowA, rowB0, rowB1, wgt, momb, k0, kend, acc0, acc1, bacc);
        else
            gram_loop<false, 0>(rowA, rowB0, rowB1, wgt, momb, k0, kend, acc0, acc1, bacc);
    } else {
        const float* invm = ws + WS_WGT + (size_t)b * Natoms;
        if ((wave & 1) == 0)
            gram_loop_compact<true>(rowA, rowB0, rowB1, invm, momb, hofs, k0, kend,
                                    acc0, acc1, bacc);
        else
            gram_loop_compact<false>(rowA, rowB0, rowB1, invm, momb, hofs, k0, kend,
                                     acc0, acc1, bacc);
    }

    // D layout: VGPR i -> M = half*8 + i, N = l15 (per 16x16 f32 C/D table)
    float* Gb = ws + WS_G + (size_t)b * Ccon * Ccon;
    const int n0 = cj0 * 16 + l15;
    const int n1 = n0 + 16;
#pragma unroll
    for (int i = 0; i < 8; ++i) {
        int m = ci * 16 + half * 8 + i;
        atomicAdd(&Gb[m * Ccon + n0], acc0[i]);
        atomicAdd(&Gb[m * Ccon + n1], acc1[i]);
    }
    if ((wave & 1) == 0)
        atomicAdd(&ws[WS_BP + b * Ccon + ci * 16 + l15], bacc);
}

// --- kernel 3: per-batch 64x64 solve (dtm*G) lmd = b in LDS ----------------
__global__ void __launch_bounds__(64)
bme_solve_kernel(const float* __restrict__ dtm, const float* __restrict__ bme,
                 float* __restrict__ ws, float* __restrict__ out) {
    __shared__ float As[Ccon][Ccon + 1];
    __shared__ float Bs[Ccon];
    __shared__ float xs[Ccon];
    const int b = blockIdx.x;
    const int r = threadIdx.x;
    const float dt = dtm[b];
    const float* Gb = ws + WS_G + (size_t)b * Ccon * Ccon;
    for (int j = 0; j < Ccon; ++j) As[r][j] = dt * Gb[r * Ccon + j];
    Bs[r] = ws[WS_BP + b * Ccon + r];
    __syncthreads();
    for (int kk = 0; kk < Ccon - 1; ++kk) {          // SPD -> no pivoting
        if (r > kk) {
            float f = As[r][kk] / As[kk][kk];
            for (int j = kk + 1; j < Ccon; ++j) As[r][j] -= f * As[kk][j];
            Bs[r] -= f * Bs[kk];
        }
        __syncthreads();
    }
    for (int kk = Ccon - 1; kk >= 0; --kk) {         // back substitution
        if (r == kk) xs[kk] = Bs[kk] / As[kk][kk];
        __syncthreads();
        if (r < kk) Bs[r] -= As[r][kk] * xs[kk];
        __syncthreads();
    }
    ws[WS_LMD + b * Ccon + r] = xs[r];
    out[(size_t)Bsz * Kdim + b * Ccon + r] = xs[r] + bme[b * Ccon + r];
}

// --- kernel 4: mom_out = mom - dtm * (lmd^T J), b128 coalesced sweep -------
__global__ void __launch_bounds__(256)
bme_apply_kernel(const float* __restrict__ mom, const float* __restrict__ dtm,
                 const float* __restrict__ jac, const float* __restrict__ ws,
                 float* __restrict__ out) {
    int t = blockIdx.x * blockDim.x + threadIdx.x;   // over B*K/4 float4 groups
    if (t >= Bsz * Kdim / 4) return;
    const int idx4 = t * 4;
    const int b = idx4 / Kdim;                       // Kdim % 4 == 0: no straddle
    const int k = idx4 - b * Kdim;
    const float* __restrict__ Jb  = jac + (size_t)b * Ccon * Kdim + k;
    const float* __restrict__ lmd = ws + WS_LMD + b * Ccon;
    v4f acc = {};
#pragma unroll 8
    for (int c = 0; c < Ccon; ++c) {
        v4f jv = *(const v4f*)(Jb + (size_t)c * Kdim);
        acc += jv * lmd[c];
    }
    v4f m = *(const v4f*)(mom + idx4);
    v4f o = m - acc * dtm[b];
    *(v4f*)(out + idx4) = o;
}

extern "C" void kernel_launch(void* const* d_in, const int* in_sizes, int n_in,
                              void* d_out, int out_size, void* d_ws, size_t ws_size,
                              hipStream_t stream) {
    const float* mom = (const float*)d_in[0];   // [B,N,3]
    const float* mas = (const float*)d_in[1];   // [B,N]
    const float* dtm = (const float*)d_in[2];   // [B]
    const float* jac = (const float*)d_in[3];   // [B,C,N,3]
    const float* bme = (const float*)d_in[4];   // [B,C]
    float* out = (float*)d_out;                 // [B,N,3] ++ [B,C]
    float* ws  = (float*)d_ws;
    (void)in_sizes; (void)n_in; (void)out_size;

    const size_t need_expanded = (size_t)(WS_HDR + (size_t)Bsz * Kdim) * sizeof(float);
    const int init_blocks = (Bsz * Natoms + 255) / 256;

    if (ws_size >= need_expanded) {
        bme_init_expand<<<init_blocks, 256, 0, stream>>>(mas, ws);
        bme_gram_kernel<0><<<Bsz * CHUNKS, 256, 0, stream>>>(jac, mom, ws);
    } else {
        bme_init_compact<<<init_blocks, 256, 0, stream>>>(mas, ws);
        bme_gram_kernel<1><<<Bsz * CHUNKS, 256, 0, stream>>>(jac, mom, ws);
    }
    bme_solve_kernel<<<Bsz, 64, 0, stream>>>(dtm, bme, ws, out);
    bme_apply_kernel<<<(Bsz * Kdim / 4 + 255) / 256, 256, 0, stream>>>(mom, dtm, jac, ws, out);
}